// N3Tree_6528350290563
// MI455X (gfx1250) — compile-verified
//
#include <hip/hip_runtime.h>
#include <hip/hip_bf16.h>
#include <stdint.h>

// N3Tree vertical query for MI455X (gfx1250, wave32).
// Latency-bound octree descent: stage hot top-of-tree child table into LDS
// via the CDNA5 Tensor Data Mover, then run 4 interleaved descent chains per
// lane to keep multiple dependent gathers in flight. Staged lookups use an
// explicit ds_load_b32 so the compiler cannot fold the LDS/global choice into
// a generic-pointer select (flat_load).

#define TREE_N        2
#define DATA_DIM      4
#define TREE_DEPTH    6
#define BLOCK         256
#define PTS_PER_THR   4
#define MAX_STAGE     1024   // nodes staged in LDS: 1024 * 8 * 4B = 32 KB

typedef unsigned int u32x4 __attribute__((ext_vector_type(4)));
typedef int          i32x8 __attribute__((ext_vector_type(8)));
typedef int          i32x4 __attribute__((ext_vector_type(4)));

struct TdmDesc {
    unsigned int g0[4];  // D# group 0 (128b)
    int          g1[8];  // D# group 1 (256b)
};

struct F3 { float x, y, z; };  // 12B, loads as global_load_b96

__global__ __launch_bounds__(BLOCK) void n3tree_query_kernel(
    const float* __restrict__ data,      // [total, 2,2,2, 4] f32
    const int*   __restrict__ child,     // [total, 2,2,2]    i32
    const F3*    __restrict__ indices,   // [Q]               f32x3
    const float* __restrict__ offset,    // [3]
    const float* __restrict__ invradius, // [1]
    float*       __restrict__ out,       // [Q, 4]
    int Q, int staged_nodes, TdmDesc desc)
{
    __shared__ int s_child[MAX_STAGE * 8];   // LDS offset 0 (only LDS object)

    // ---- Stage top of tree into LDS ---------------------------------------
#if defined(__has_builtin)
#if __has_builtin(__builtin_amdgcn_tensor_load_to_lds) && \
    __has_builtin(__builtin_amdgcn_s_wait_tensorcnt)
#define USE_TDM_STAGE 1
#endif
#endif
#ifdef USE_TDM_STAGE
    if (threadIdx.x < 32u) {               // wave 0 only; EXEC ignored by TDM
        u32x4 g0 = { desc.g0[0], desc.g0[1], desc.g0[2], desc.g0[3] };
        i32x8 g1 = { desc.g1[0], desc.g1[1], desc.g1[2], desc.g1[3],
                     desc.g1[4], desc.g1[5], desc.g1[6], desc.g1[7] };
        i32x4 gz4 = { 0, 0, 0, 0 };        // groups 2/3 unused (<=2D tensor)
        i32x8 gz8 = { 0, 0, 0, 0, 0, 0, 0, 0 };
        // clang-23 / therock-10.0 form: 6 args (extra int32x8 before cpol)
        __builtin_amdgcn_tensor_load_to_lds(g0, g1, gz4, gz4, gz8, 0);
        __builtin_amdgcn_s_wait_tensorcnt(0);
    }
#else
    for (int i = threadIdx.x; i < staged_nodes * 8; i += BLOCK)
        s_child[i] = child[i];
#endif
    __syncthreads();

    // Keep the LDS allocation alive: the staged reads below go through raw
    // ds_load asm (byte offsets from LDS base 0), so s_child is otherwise
    // unreferenced and the compiler would drop its allocation — which would
    // make the TDM write target unallocated LDS.
    asm volatile("" : : "r"(&s_child[0]) : "memory");

    const float invr = invradius[0];
    const float ox = offset[0], oy = offset[1], oz = offset[2];
    const float4* __restrict__ data4 = (const float4*)data;

    const int base = blockIdx.x * (BLOCK * PTS_PER_THR) + threadIdx.x;

    int    qv[PTS_PER_THR];
    float  px[PTS_PER_THR], py[PTS_PER_THR], pz[PTS_PER_THR];
    int    node[PTS_PER_THR];
    bool   act[PTS_PER_THR];
    float4 res[PTS_PER_THR];

#pragma unroll
    for (int k = 0; k < PTS_PER_THR; ++k) {
        qv[k] = base + k * BLOCK;
        F3 v = { 0.f, 0.f, 0.f };
        if (qv[k] < Q) v = indices[qv[k]];
        px[k] = fminf(fmaxf(ox + v.x * invr, 0.f), 1.0f);
        py[k] = fminf(fmaxf(oy + v.y * invr, 0.f), 1.0f);
        pz[k] = fminf(fmaxf(oz + v.z * invr, 0.f), 1.0f);
        node[k] = 0;
        act[k]  = true;
        res[k]  = make_float4(0.f, 0.f, 0.f, 0.f);
    }

    // ---- 6-level descent, 4 chains interleaved for memory-level parallelism
#pragma unroll
    for (int l = 0; l < TREE_DEPTH; ++l) {
#pragma unroll
        for (int k = 0; k < PTS_PER_THR; ++k) {
            if (!act[k]) continue;
            px[k] *= (float)TREE_N;
            py[k] *= (float)TREE_N;
            pz[k] *= (float)TREE_N;
            float fxf = fminf(floorf(px[k]), (float)(TREE_N - 1));
            float fyf = fminf(floorf(py[k]), (float)(TREE_N - 1));
            float fzf = fminf(floorf(pz[k]), (float)(TREE_N - 1));
            px[k] -= fxf; py[k] -= fyf; pz[k] -= fzf;
            int fx = (int)fxf, fy = (int)fyf, fz = (int)fzf;
            int cidx = node[k] * 8 + (fx << 2) + (fy << 1) + fz;

            int delta;
            if (node[k] < staged_nodes) {
                // Staged top of tree: dedicated LDS pipe, byte addr = cidx*4
                // (s_child starts at LDS offset 0, where the TDM wrote it).
                int baddr = cidx << 2;
                asm volatile("ds_load_b32 %0, %1\n\t"
                             "s_wait_dscnt 0"
                             : "=v"(delta) : "v"(baddr));
            } else {
                delta = child[cidx];                               // global_load_b32
            }

            if (delta == 0) {
                res[k] = data4[cidx];                              // global_load_b128
                act[k] = false;
            } else {
                node[k] += delta;
            }
        }
    }

#pragma unroll
    for (int k = 0; k < PTS_PER_THR; ++k) {
        if (qv[k] < Q)
            *(float4*)(out + 4 * (size_t)qv[k]) = res[k];          // global_store_b128
    }
}

extern "C" void kernel_launch(void* const* d_in, const int* in_sizes, int n_in,
                              void* d_out, int out_size, void* d_ws, size_t ws_size,
                              hipStream_t stream) {
    const float* data      = (const float*)d_in[0];
    const int*   child     = (const int*)  d_in[1];
    const F3*    indices   = (const F3*)   d_in[2];
    const float* offset    = (const float*)d_in[3];
    const float* invradius = (const float*)d_in[4];
    float*       out       = (float*)d_out;

    const int Q     = in_sizes[2] / 3;
    const int total = in_sizes[1] / 8;
    const int staged = total < MAX_STAGE ? total : MAX_STAGE;

    // ---- Build TDM descriptor on the host (device VA known here) ----------
    // D# group 0: count=1 | lds_addr=0 | global_addr=&child[0] | type=2
    // D# group 1: data_size=4B; 1-row tile of K dwords (K = staged*8 <= 8192)
    TdmDesc desc = {};
    const unsigned long long ga = (unsigned long long)(uintptr_t)child;
    const unsigned int K = (unsigned int)(staged * 8);  // dwords to copy

    desc.g0[0] = 0x1u;                                   // count=1, user mode
    desc.g0[1] = 0u;                                     // lds_addr = 0
    desc.g0[2] = (unsigned int)(ga & 0xffffffffu);       // global_addr[31:0]
    desc.g0[3] = (unsigned int)((ga >> 32) & 0x01ffffffu)// global_addr[56:32]
               | 0x80000000u;                            // type = 2 ("image")

    desc.g1[0] = (int)(2u << 16);                        // data_size = 2 (4B)
    desc.g1[1] = (int)((K & 0xffffu) << 16);             // tensor_dim0[15:0] @bit48
    desc.g1[2] = (int)(((K >> 16) & 0xffffu)             // tensor_dim0[31:16]
               | (1u << 16));                            // tensor_dim1 = 1
    desc.g1[3] = (int)((K & 0xffffu) << 16);             // tile_dim0 = K @bit112
    desc.g1[4] = 1;                                      // tile_dim1=1, tile_dim2=0
    desc.g1[5] = (int)K;                                 // tensor_dim0_stride[31:0]
    desc.g1[6] = 0;                                      // stride0 hi | stride1 lo
    desc.g1[7] = 0;

    const int pts_per_block = BLOCK * PTS_PER_THR;
    const int grid = (Q + pts_per_block - 1) / pts_per_block;

    n3tree_query_kernel<<<grid, BLOCK, 0, stream>>>(
        data, child, indices, offset, invradius, out, Q, staged, desc);
}